// pointer_network_27805618274770
// MI455X (gfx1250) — compile-verified
//
#include <hip/hip_runtime.h>
#include <hip/hip_bf16.h>
#include <math.h>

#define Bsz 256
#define Ssz 512
#define Dsz 64
#define Hsz 256
#define Wsz 256
#define SP1 513
#define MAXDEC 16

typedef float v2f __attribute__((ext_vector_type(2)));
typedef float v8f __attribute__((ext_vector_type(8)));

// ---------------- device-global state (avoids ws_size assumptions) ----------------
__device__ float g_Enc[(size_t)SP1 * Bsz * Hsz];   // [t][b][h]  encoder states
__device__ float g_W1E[(size_t)SP1 * Bsz * Wsz];   // [t][b][w]  hoisted W1 @ Enc
__device__ float g_mask[(size_t)Bsz * SP1];
__device__ float g_term[Bsz];
__device__ float g_logp[Bsz];
__device__ float g_hidden[Bsz * Hsz];
__device__ float g_lin[Bsz * Hsz];
__device__ float g_aff[Bsz * Wsz];
__device__ float g_dec_h[Bsz * Hsz];
__device__ float g_dec_in[Bsz * Dsz];
__device__ int   g_sched[MAXDEC * Bsz];

// ---------------- fp32 WMMA: D(16x16) += A(16x4) * B(4x16), wave32 ----------------
__device__ __forceinline__ v8f wmma4(v2f a, v2f b, v8f c) {
  return __builtin_amdgcn_wmma_f32_16x16x4_f32(
      /*neg_a=*/false, a, /*neg_b=*/false, b,
      /*c_mod=*/(short)0, c, /*reuse_a=*/false, /*reuse_b=*/false);
}

__device__ __forceinline__ float sigm(float x) { return 1.0f / (1.0f + expf(-x)); }

// ---------------- fused GRU gate tile: out = GRUCell(x, h) for one 16x16 tile -----
// Weights are torch-layout [3H, K] row-major (gemm is x @ W^T).
__device__ void gru_tile(const float* xbase, size_t xstride, int xvalid, int Kx,
                         const float* hbase, int hvalid,
                         const float* __restrict__ Wih, const float* __restrict__ Whh,
                         const float* __restrict__ bih, const float* __restrict__ bhh,
                         float* __restrict__ out)
{
  const int lane = threadIdx.x & 31;
  const int m  = lane & 15;          // A row / C col index
  const int kh = lane >> 4;          // K half select
  const int n  = m;                  // B col (lanes 0-15 carry N)
  const int row0 = blockIdx.x * 16;                       // batch tile
  const int col0 = (blockIdx.y * 4 + threadIdx.y) * 16;   // hidden tile

  v8f ar = {}, az = {}, ain = {}, ahn = {};

  if (xvalid) {
    const float* xr = xbase + (size_t)(row0 + m) * xstride;
    const float* wr = Wih + (size_t)(col0 + n) * Kx;
    const float* wz = Wih + (size_t)(Hsz + col0 + n) * Kx;
    const float* wn = Wih + (size_t)(2 * Hsz + col0 + n) * Kx;
    for (int kb = 0; kb < Kx; kb += 4) {
      const int kk = kb + 2 * kh;
      v2f a;  a.x = xr[kk];  a.y = xr[kk + 1];
      v2f b1; b1.x = wr[kk]; b1.y = wr[kk + 1];
      ar = wmma4(a, b1, ar);
      v2f b2; b2.x = wz[kk]; b2.y = wz[kk + 1];
      az = wmma4(a, b2, az);
      v2f b3; b3.x = wn[kk]; b3.y = wn[kk + 1];
      ain = wmma4(a, b3, ain);
    }
  }
  if (hvalid) {
    const float* hr = hbase + (size_t)(row0 + m) * Hsz;
    const float* wr = Whh + (size_t)(col0 + n) * Hsz;
    const float* wz = Whh + (size_t)(Hsz + col0 + n) * Hsz;
    const float* wn = Whh + (size_t)(2 * Hsz + col0 + n) * Hsz;
    for (int kb = 0; kb < Hsz; kb += 4) {
      const int kk = kb + 2 * kh;
      v2f a;  a.x = hr[kk];  a.y = hr[kk + 1];
      v2f b1; b1.x = wr[kk]; b1.y = wr[kk + 1];
      ar = wmma4(a, b1, ar);
      v2f b2; b2.x = wz[kk]; b2.y = wz[kk + 1];
      az = wmma4(a, b2, az);
      v2f b3; b3.x = wn[kk]; b3.y = wn[kk + 1];
      ahn = wmma4(a, b3, ahn);
    }
  }

  const float br_  = bih[col0 + n] + bhh[col0 + n];
  const float bz_  = bih[Hsz + col0 + n] + bhh[Hsz + col0 + n];
  const float bin_ = bih[2 * Hsz + col0 + n];
  const float bhn_ = bhh[2 * Hsz + col0 + n];
  for (int r = 0; r < 8; ++r) {
    const int mr = row0 + r + kh * 8;
    const float hv = hvalid ? hbase[(size_t)mr * Hsz + col0 + n] : 0.0f;
    const float rg = sigm(ar[r] + br_);
    const float zg = sigm(az[r] + bz_);
    const float ng = tanhf(ain[r] + bin_ + rg * (ahn[r] + bhn_));
    out[(size_t)mr * Hsz + col0 + n] = (1.0f - zg) * ng + zg * hv;
  }
}

// ---------------- encoder scan: one launch per t, stream-ordered -------------------
__global__ __launch_bounds__(128)
void enc_step_kernel(const float* __restrict__ fm,
                     const float* __restrict__ Wih, const float* __restrict__ Whh,
                     const float* __restrict__ bih, const float* __restrict__ bhh,
                     int t)
{
  const float* xb = (t > 0) ? (fm + (size_t)(t - 1) * Dsz) : nullptr;
  const float* hb = (t > 0) ? (g_Enc + (size_t)(t - 1) * Bsz * Hsz) : nullptr;
  float* out = g_Enc + (size_t)t * Bsz * Hsz;
  gru_tile(xb, (size_t)Ssz * Dsz, t > 0, Dsz, hb, t > 0, Wih, Whh, bih, bhh, out);
}

__global__ __launch_bounds__(128)
void dec_gates_kernel(const float* __restrict__ Wih, const float* __restrict__ Whh,
                      const float* __restrict__ bih, const float* __restrict__ bhh)
{
  gru_tile(g_dec_in, Dsz, 1, Dsz, g_dec_h, 1, Wih, Whh, bih, bhh, g_hidden);
}

// ---------------- generic C = [A|A2] @ W^T (+bias, tanh, *term), fp32 WMMA ---------
// mode 0: W1E = Enc @ W1^T         mode 1: aff = dec_h @ W2^T
// mode 2: dec_h = tanh([hidden|lin] @ Wa^T + ba) * term
__global__ __launch_bounds__(128)
void gemm_nt_kernel(const float* __restrict__ Wt, const float* __restrict__ bias, int mode)
{
  const float* A;  const float* A2 = nullptr;  float* C;
  size_t lda, ldc; int K1, K2 = 0, ldw, act = 0, useterm = 0;
  if (mode == 0)      { A = g_Enc;    C = g_W1E;   lda = Hsz; ldc = Wsz; K1 = Hsz; ldw = Hsz; }
  else if (mode == 1) { A = g_dec_h;  C = g_aff;   lda = Hsz; ldc = Wsz; K1 = Hsz; ldw = Hsz; }
  else                { A = g_hidden; A2 = g_lin;  C = g_dec_h; lda = Hsz; ldc = Hsz;
                        K1 = Hsz; K2 = Hsz; ldw = 2 * Hsz; act = 1; useterm = 1; }

  const int lane = threadIdx.x & 31;
  const int m = lane & 15, kh = lane >> 4, n = m;
  const int row0 = blockIdx.x * 16;
  const int col0 = (blockIdx.y * 4 + threadIdx.y) * 16;

  v8f acc = {};
  const float* arow = A + (size_t)(row0 + m) * lda;
  const float* wrow = Wt + (size_t)(col0 + n) * ldw;
  for (int kb = 0; kb < K1; kb += 4) {
    const int kk = kb + 2 * kh;
    v2f a; a.x = arow[kk]; a.y = arow[kk + 1];
    v2f b; b.x = wrow[kk]; b.y = wrow[kk + 1];
    acc = wmma4(a, b, acc);
  }
  if (K2 > 0) {
    const float* arow2 = A2 + (size_t)(row0 + m) * lda;
    for (int kb = 0; kb < K2; kb += 4) {
      const int kk = kb + 2 * kh;
      v2f a; a.x = arow2[kk]; a.y = arow2[kk + 1];
      v2f b; b.x = wrow[K1 + kk]; b.y = wrow[K1 + kk + 1];
      acc = wmma4(a, b, acc);
    }
  }
  const float bv = bias ? bias[col0 + n] : 0.0f;
  for (int r = 0; r < 8; ++r) {
    const int mr = row0 + r + kh * 8;
    float v = acc[r] + bv;
    if (act) v = tanhf(v);
    if (useterm) v *= g_term[mr];
    C[(size_t)mr * ldc + col0 + n] = v;
  }
}

// ---------------- attention scores + softmax + pointer selection (1 block / b) -----
__global__ __launch_bounds__(256)
void attn_select_kernel(const float* __restrict__ fm, const float* __restrict__ V, int step)
{
  const int b = blockIdx.x;
  const int tid = threadIdx.x;
  __shared__ float s_aff[Wsz];
  __shared__ float s_prob[SP1];
  __shared__ float s_masked[SP1];
  __shared__ float s_red[256];
  __shared__ int   s_idx;
  __shared__ float s_termv;

  s_aff[tid] = g_aff[b * Wsz + tid];
  __syncthreads();

  // scores[s] = V . tanh(W1E[s,b,:] + aff[b,:])
  for (int s = tid; s < SP1; s += 256) {
    const float* we = &g_W1E[((size_t)s * Bsz + b) * Wsz];
    float acc = 0.0f;
    for (int w = 0; w < Wsz; ++w) acc += V[w] * tanhf(we[w] + s_aff[w]);
    s_prob[s] = acc;
  }
  __syncthreads();

  // softmax (TEMP = 1.0)
  float lm = -3.4e38f;
  for (int s = tid; s < SP1; s += 256) lm = fmaxf(lm, s_prob[s]);
  s_red[tid] = lm; __syncthreads();
  for (int off = 128; off > 0; off >>= 1) {
    if (tid < off) s_red[tid] = fmaxf(s_red[tid], s_red[tid + off]);
    __syncthreads();
  }
  const float mx = s_red[0]; __syncthreads();
  float ls = 0.0f;
  for (int s = tid; s < SP1; s += 256) { float e = expf(s_prob[s] - mx); s_prob[s] = e; ls += e; }
  s_red[tid] = ls; __syncthreads();
  for (int off = 128; off > 0; off >>= 1) {
    if (tid < off) s_red[tid] += s_red[tid + off];
    __syncthreads();
  }
  const float inv = 1.0f / s_red[0]; __syncthreads();
  for (int s = tid; s < SP1; s += 256) {
    const float p = s_prob[s] * inv;
    s_prob[s] = p;
    s_masked[s] = p * g_mask[(size_t)b * SP1 + s];
  }
  __syncthreads();

  if (tid == 0) {
    float b1 = -1.0f; int i1 = 0; float b2 = -1.0f; int i2 = 0;
    for (int s = 0; s < SP1; ++s) {
      const float v = s_masked[s];
      if (v > b1) { b2 = b1; i2 = i1; b1 = v; i1 = s; }
      else if (v > b2) { b2 = v; i2 = s; }
    }
    int idx = i1;
    if (step < 2 && idx == 0) idx = i2;          // MIN_DEC rule
    const float ip = s_masked[idx];               // index_probs (pre mask0-fix)
    const float m0 = g_mask[(size_t)b * SP1];
    if (m0 == 0.0f) idx = 0;
    const float tnew = g_term[b] * m0;            // term uses pre-update mask[:,0]
    g_term[b] = tnew;
    g_logp[b] += tnew * logf(ip + 1e-9f);
    g_mask[(size_t)b * SP1 + idx] = 0.0f;         // mask *= (1 - one_hot)
    g_sched[step * Bsz + b] = idx;
    s_idx = idx; s_termv = tnew;
  }
  __syncthreads();
  const int idx = s_idx; const float tm = s_termv;

  // dec_in_next = aug[b, idx] * term   (aug row 0 is zeros)
  if (tid < Dsz) {
    const float sv = (idx == 0) ? 0.0f : fm[((size_t)b * Ssz + (idx - 1)) * Dsz + tid];
    g_dec_in[b * Dsz + tid] = sv * tm;
  }
  // lin[b,h] = sum_s prob[s] * Enc[s,b,h]
  {
    const int h = tid;
    float acc = 0.0f;
    for (int s = 0; s < SP1; ++s) acc += s_prob[s] * g_Enc[((size_t)s * Bsz + b) * Hsz + h];
    g_lin[b * Hsz + h] = acc;
  }
}

// ---------------- init / hT copy / output --------------------------------------
__global__ void init_kernel(const float* __restrict__ dec_in0)
{
  const int i = blockIdx.x * blockDim.x + threadIdx.x;
  if (i < Bsz * SP1) g_mask[i] = 1.0f;
  if (i < Bsz) { g_term[i] = 1.0f; g_logp[i] = 0.0f; }
  if (i < Bsz * Dsz) g_dec_in[i] = dec_in0[i % Dsz];
}

__global__ void copy_hT_kernel()
{
  const int i = blockIdx.x * blockDim.x + threadIdx.x;
  if (i < Bsz * Hsz) g_dec_h[i] = g_Enc[(size_t)(SP1 - 1) * Bsz * Hsz + i];
}

__global__ void finalize_kernel(float* __restrict__ out)
{
  const int i = blockIdx.x * blockDim.x + threadIdx.x;
  if (i < Bsz) out[i] = g_logp[i];
  if (i < Bsz * MAXDEC) {
    const int b = i / MAXDEC, st = i % MAXDEC;
    out[Bsz + b * MAXDEC + st] = (float)g_sched[st * Bsz + b];
  }
}

// ---------------- host orchestration -------------------------------------------
extern "C" void kernel_launch(void* const* d_in, const int* in_sizes, int n_in,
                              void* d_out, int out_size, void* d_ws, size_t ws_size,
                              hipStream_t stream)
{
  (void)in_sizes; (void)n_in; (void)d_ws; (void)ws_size; (void)out_size;
  const float* fm      = (const float*)d_in[0];
  const float* eWih    = (const float*)d_in[1];
  const float* eWhh    = (const float*)d_in[2];
  const float* ebih    = (const float*)d_in[3];
  const float* ebhh    = (const float*)d_in[4];
  const float* dWih    = (const float*)d_in[5];
  const float* dWhh    = (const float*)d_in[6];
  const float* dbih    = (const float*)d_in[7];
  const float* dbhh    = (const float*)d_in[8];
  const float* W1      = (const float*)d_in[9];
  const float* W2      = (const float*)d_in[10];
  const float* V       = (const float*)d_in[11];
  const float* Wa      = (const float*)d_in[12];
  const float* ba      = (const float*)d_in[13];
  const float* dec_in0 = (const float*)d_in[14];
  float* out = (float*)d_out;

  const dim3 blk(32, 4);   // 4 waves, each owns a 16x16 output tile

  init_kernel<<<(Bsz * SP1 + 255) / 256, 256, 0, stream>>>(dec_in0);

  // encoder scan: 513 stream-ordered steps (weights are L2-resident; latency-bound)
  for (int t = 0; t < SP1; ++t)
    enc_step_kernel<<<dim3(Bsz / 16, Hsz / 64), blk, 0, stream>>>(fm, eWih, eWhh, ebih, ebhh, t);

  copy_hT_kernel<<<(Bsz * Hsz + 255) / 256, 256, 0, stream>>>();

  // hoisted W1E = Enc @ W1^T : 131328 x 256 x 256, the throughput GEMM
  gemm_nt_kernel<<<dim3((SP1 * Bsz) / 16, Wsz / 64), blk, 0, stream>>>(W1, nullptr, 0);

  for (int i = 0; i < MAXDEC; ++i) {
    dec_gates_kernel<<<dim3(Bsz / 16, Hsz / 64), blk, 0, stream>>>(dWih, dWhh, dbih, dbhh);
    gemm_nt_kernel<<<dim3(Bsz / 16, Wsz / 64), blk, 0, stream>>>(W2, nullptr, 1);   // aff = dec_h @ W2^T
    attn_select_kernel<<<Bsz, 256, 0, stream>>>(fm, V, i);
    gemm_nt_kernel<<<dim3(Bsz / 16, Hsz / 64), blk, 0, stream>>>(Wa, ba, 2);        // dec_h = tanh(...)*term
  }

  finalize_kernel<<<(Bsz * MAXDEC + 255) / 256, 256, 0, stream>>>(out);
}